// HKRPQLinear_17523466567937
// MI455X (gfx1250) — compile-verified
//
#include <hip/hip_runtime.h>

typedef float v2f __attribute__((ext_vector_type(2)));
typedef float v8f __attribute__((ext_vector_type(8)));

#define IN_F   4096
#define OUT_F  4096
#define NROWS  8192
#define NC     32      // number of codebooks (C)
#define NCL    32      // number of clusters
#define SUBD   128     // sub-dim per codebook
#define BM     128
#define BN     128
#define BK     32
#define NKT    (IN_F / BK)   // 128 K-tiles
#define LDT    36      // padded LDS row stride (floats): 16B-aligned, bank-tiling friendly

// ---------------- kernel 1: expand centroid weights, transposed: cwT[k][n] ----------------
__global__ void expand_cwT(const float* __restrict__ codebooks,   // [C][256][128]
                           const int*   __restrict__ centroidsT,  // [C][NCL]
                           float* __restrict__ cwT)                // [IN_F][NCL]
{
    int idx = blockIdx.x * blockDim.x + threadIdx.x;   // idx = k*NCL + n
    if (idx >= IN_F * NCL) return;
    int n = idx & (NCL - 1);
    int k = idx >> 5;
    int c = k >> 7;        // k / 128
    int s = k & 127;
    int e = centroidsT[c * NCL + n];
    cwT[idx] = codebooks[((size_t)(c * 256 + e) * SUBD) + s];
}

// ---------------- kernel 2: routing dots + LN + softmax + masks (wave per row) ----------------
__global__ __launch_bounds__(256)
void dots_masks(const float* __restrict__ x,         // [NROWS][IN_F]
                const float* __restrict__ cwT,       // [IN_F][NCL]
                const float* __restrict__ ln_weight, // [NCL]
                unsigned* __restrict__ qmask,        // [NROWS]
                unsigned* __restrict__ cmask)        // [NCL]
{
    int wave = threadIdx.x >> 5;
    int lane = threadIdx.x & 31;
    int row  = blockIdx.x * 8 + wave;

    const float* xr = x + (size_t)row * IN_F;
    float acc = 0.f;
    for (int k = 0; k < IN_F; k += 4) {
        float4 xv = *(const float4*)(xr + k);
        acc = fmaf(xv.x, cwT[(k + 0) * NCL + lane], acc);
        acc = fmaf(xv.y, cwT[(k + 1) * NCL + lane], acc);
        acc = fmaf(xv.z, cwT[(k + 2) * NCL + lane], acc);
        acc = fmaf(xv.w, cwT[(k + 3) * NCL + lane], acc);
    }
    // mean over 32 lanes
    float s = acc;
    for (int off = 16; off >= 1; off >>= 1) s += __shfl_xor(s, off, 32);
    float mu = s * (1.f / 32.f);
    float d  = acc - mu;
    float v  = d * d;
    for (int off = 16; off >= 1; off >>= 1) v += __shfl_xor(v, off, 32);
    v *= (1.f / 32.f);
    float logit = d * rsqrtf(v + 1e-5f) * ln_weight[lane];
    // softmax (max-stabilized)
    float mx = logit;
    for (int off = 16; off >= 1; off >>= 1) mx = fmaxf(mx, __shfl_xor(mx, off, 32));
    float e = expf(logit - mx);
    float se = e;
    for (int off = 16; off >= 1; off >>= 1) se += __shfl_xor(se, off, 32);
    float p = e / se;
    bool over = p > 0.5f;
    unsigned long long bal = __ballot(over);
    if (lane == 0) qmask[row] = (bal != 0ull) ? 1u : 0u;
    if (over) atomicOr(&cmask[lane], 1u);
}

// ---------------- kernel 3: main GEMM, double-buffered LDS + register-staged prefetch ----------------
__global__ __launch_bounds__(256)
void pq_gemm(const float* __restrict__ x,          // [NROWS][IN_F]
             const float* __restrict__ codebooks,  // [C][256][128]
             const int*   __restrict__ codesT,     // [C][OUT_F]
             const float* __restrict__ bias,       // [OUT_F]
             const unsigned* __restrict__ qmask,   // [NROWS]
             const unsigned* __restrict__ cmask,   // [NCL]
             float* __restrict__ out)              // [NROWS][OUT_F]
{
    __shared__ float lA[2][BM * LDT];
    __shared__ float lB[2][BN * LDT];

    const int tid  = threadIdx.x;
    const int wave = tid >> 5;
    const int lane = tid & 31;
    const int m0 = blockIdx.y * BM;
    const int n0 = blockIdx.x * BN;
    const int wm = (wave & 3) * 32;   // wave M-offset inside block tile (4 waves over M)
    const int wn = (wave >> 2) * 64;  // wave N-offset inside block tile (2 waves over N)
    const int rl = lane & 15;
    const int khalf = (lane >> 4) << 1;   // lanes 0-15 -> K {0,1}; lanes 16-31 -> K {2,3}

    // per-thread staging geometry: 4 float4 slots per tile, slot it covers
    // flat float4 index f = it*256 + tid; row = f>>3 (0..127); col = (f&7)*4 (0..28)
    int srow[4], scol[4];
#pragma unroll
    for (int it = 0; it < 4; ++it) {
        int f = it * 256 + tid;
        srow[it] = f >> 3;
        scol[it] = (f & 7) << 2;
    }

    v8f acc[2][4];
#pragma unroll
    for (int i = 0; i < 2; ++i)
#pragma unroll
        for (int j = 0; j < 4; ++j)
            acc[i][j] = {};

    float4 ra[4], rb[4];

    // ---- global -> registers for K-tile kt ----
    auto stage_load = [&](int k0) {
        int cb = k0 >> 7;      // codebook index (BK | SUBD -> single codebook per tile)
        int ks = k0 & 127;
        const float* cbase = codebooks + (size_t)cb * 256 * SUBD + ks;
        const int*   crow  = codesT + cb * OUT_F + n0;
#pragma unroll
        for (int it = 0; it < 4; ++it) {
            ra[it] = *(const float4*)(x + (size_t)(m0 + srow[it]) * IN_F + k0 + scol[it]);
            int e  = crow[srow[it]];
            rb[it] = *(const float4*)(cbase + (size_t)e * SUBD + scol[it]);
        }
    };
    // ---- registers -> LDS buffer b ----
    auto stage_store = [&](int b) {
#pragma unroll
        for (int it = 0; it < 4; ++it) {
            *(float4*)&lA[b][srow[it] * LDT + scol[it]] = ra[it];
            *(float4*)&lB[b][srow[it] * LDT + scol[it]] = rb[it];
        }
    };

    // prologue: tile 0 into buffer 0
    stage_load(0);
    stage_store(0);
    __syncthreads();

    for (int kt = 0; kt < NKT; ++kt) {
        // issue next tile's global loads before compute so they overlap the WMMAs
        if (kt + 1 < NKT) stage_load((kt + 1) * BK);

        const float* A = lA[kt & 1];
        const float* B = lB[kt & 1];
#pragma unroll
        for (int k4 = 0; k4 < BK; k4 += 4) {
            int kk = k4 + khalf;
            v2f a0 = *(const v2f*)&A[(wm +      rl) * LDT + kk];
            v2f a1 = *(const v2f*)&A[(wm + 16 + rl) * LDT + kk];
            v2f b0 = *(const v2f*)&B[(wn +      rl) * LDT + kk];
            v2f b1 = *(const v2f*)&B[(wn + 16 + rl) * LDT + kk];
            v2f b2 = *(const v2f*)&B[(wn + 32 + rl) * LDT + kk];
            v2f b3 = *(const v2f*)&B[(wn + 48 + rl) * LDT + kk];

            acc[0][0] = __builtin_amdgcn_wmma_f32_16x16x4_f32(false, a0, false, b0, (short)0, acc[0][0], false, false);
            acc[0][1] = __builtin_amdgcn_wmma_f32_16x16x4_f32(false, a0, false, b1, (short)0, acc[0][1], false, false);
            acc[0][2] = __builtin_amdgcn_wmma_f32_16x16x4_f32(false, a0, false, b2, (short)0, acc[0][2], false, false);
            acc[0][3] = __builtin_amdgcn_wmma_f32_16x16x4_f32(false, a0, false, b3, (short)0, acc[0][3], false, false);
            acc[1][0] = __builtin_amdgcn_wmma_f32_16x16x4_f32(false, a1, false, b0, (short)0, acc[1][0], false, false);
            acc[1][1] = __builtin_amdgcn_wmma_f32_16x16x4_f32(false, a1, false, b1, (short)0, acc[1][1], false, false);
            acc[1][2] = __builtin_amdgcn_wmma_f32_16x16x4_f32(false, a1, false, b2, (short)0, acc[1][2], false, false);
            acc[1][3] = __builtin_amdgcn_wmma_f32_16x16x4_f32(false, a1, false, b3, (short)0, acc[1][3], false, false);
        }

        if (kt + 1 < NKT) {
            stage_store((kt + 1) & 1);  // compiler sinks s_wait_loadcnt here
            __syncthreads();            // waits dscnt, protects RAW on buffer (kt+1)&1, WAR on kt&1
        }
    }

    // ---- epilogue: bias + row/col masks ----
    // C/D layout: VGPR v, lanes 0-15 -> (M = v,   N = lane); lanes 16-31 -> (M = v+8, N = lane-16)
#pragma unroll
    for (int i = 0; i < 2; ++i) {
#pragma unroll
        for (int j = 0; j < 4; ++j) {
            int mt  = m0 + wm + i * 16 + ((lane >> 4) << 3);
            int col = n0 + wn + j * 16 + rl;
            float bcol = bias[col];
            float colm = (cmask[col >> 7] != 0u) ? 1.f : 0.f;
#pragma unroll
            for (int v = 0; v < 8; ++v) {
                int row = mt + v;
                float rowm = (qmask[row] != 0u) ? 1.f : 0.f;
                out[(size_t)row * OUT_F + col] = (acc[i][j][v] + bcol) * rowm * colm;
            }
        }
    }
}

extern "C" void kernel_launch(void* const* d_in, const int* in_sizes, int n_in,
                              void* d_out, int out_size, void* d_ws, size_t ws_size,
                              hipStream_t stream) {
    const float* x          = (const float*)d_in[0];   // (4,2048,4096) f32
    const float* codebooks  = (const float*)d_in[1];   // (32,256,128) f32
    const float* bias       = (const float*)d_in[2];   // (4096,) f32
    const float* ln_weight  = (const float*)d_in[3];   // (32,) f32
    const int*   codesT     = (const int*)d_in[4];     // (32,4096) i32
    const int*   centroidsT = (const int*)d_in[5];     // (32,32) i32
    float* out = (float*)d_out;

    // workspace layout
    float*    cwT   = (float*)d_ws;                    // 4096*32 f32 = 512 KB
    unsigned* qmask = (unsigned*)(cwT + IN_F * NCL);   // 8192 u32
    unsigned* cmask = qmask + NROWS;                   // 32 u32

    hipMemsetAsync(cmask, 0, NCL * sizeof(unsigned), stream);

    expand_cwT<<<(IN_F * NCL) / 256, 256, 0, stream>>>(codebooks, centroidsT, cwT);
    dots_masks<<<NROWS / 8, 256, 0, stream>>>(x, cwT, ln_weight, qmask, cmask);

    dim3 grid(OUT_F / BN, NROWS / BM);
    pq_gemm<<<grid, 256, 0, stream>>>(x, codebooks, codesT, bias, qmask, cmask, out);
}